// GetScalars_72138270703994
// MI455X (gfx1250) — compile-verified
//
#include <hip/hip_runtime.h>

// SO(3) invariant "GetScalars": per point (b,n) compute l=0 channel interleave
// plus per-l 16x16 complex Gram matrices over the m dimension, via
// V_WMMA_F32_16X16X4_F32 (one wave32 per point).

typedef float v2f __attribute__((ext_vector_type(2)));
typedef float v4f __attribute__((ext_vector_type(4)));
typedef float v8f __attribute__((ext_vector_type(8)));

#define NCH 16           // channels C
#define OUT_STRIDE 2112  // 1056 complex64 per point, in floats

// Load a 4(m) x 16(c) f32 tile in the shared A/B register image for
// v_wmma_f32_16x16x4_f32:
//   VGPR j, lane L:  tile[m0 + 2*(L>=16) + j,  L&15]   (zero-padded past NM)
// This image is simultaneously A = tile^T (16x4) and B = tile (4x16).
template <int NM>
__device__ __forceinline__ v2f load_tile(const float* __restrict__ base, int m0, int lane) {
  const int c  = lane & 15;
  const int mh = (lane >> 4) << 1;  // 0 for lanes 0-15, 2 for lanes 16-31
  const int mx = m0 + mh;
  const int my = mx + 1;
  v2f t;
  t.x = (mx < NM) ? base[mx * NCH + c] : 0.0f;
  t.y = (my < NM) ? base[my * NCH + c] : 0.0f;
  return t;
}

// One degree l: G_re = Re^T Re + Im^T Im ; G_im = Im^T Re - Re^T Im.
// D layout: lane L, vgpr v -> row c1 = v + 8*(L>=16), col c2 = L&15.
template <int NM>
__device__ __forceinline__ void gram_l(const float* __restrict__ re,
                                       const float* __restrict__ im,
                                       float* __restrict__ obase, int lane) {
  v8f gre = {0.f, 0.f, 0.f, 0.f, 0.f, 0.f, 0.f, 0.f};
  v8f gim = {0.f, 0.f, 0.f, 0.f, 0.f, 0.f, 0.f, 0.f};
#pragma unroll
  for (int m0 = 0; m0 < NM; m0 += 4) {
    v2f tre = load_tile<NM>(re, m0, lane);
    v2f tim = load_tile<NM>(im, m0, lane);
    v2f tnr;                     // f32 WMMA has no A/B neg modifier: negate Re tile in VALU
    tnr.x = -tre.x;
    tnr.y = -tre.y;
    gre = __builtin_amdgcn_wmma_f32_16x16x4_f32(false, tre, false, tre, (short)0, gre, false, false);
    gre = __builtin_amdgcn_wmma_f32_16x16x4_f32(false, tim, false, tim, (short)0, gre, false, false);
    gim = __builtin_amdgcn_wmma_f32_16x16x4_f32(false, tim, false, tre, (short)0, gim, false, false);
    gim = __builtin_amdgcn_wmma_f32_16x16x4_f32(false, tnr, false, tim, (short)0, gim, false, false);
  }
  const int hi = lane >> 4;
  const int c2 = lane & 15;
#pragma unroll
  for (int v = 0; v < 8; ++v) {
    const int c1 = v + 8 * hi;
    v2f val;
    val.x = gre[v];
    val.y = gim[v];
    // complex64 element (g_re, g_im) at flat index c1*16 + c2; streamed, never re-read
    __builtin_nontemporal_store(val, (v2f*)(obase + (c1 * 16 + c2) * 2));
  }
}

__global__ __launch_bounds__(256) void so3_scalars_kernel(
    const float* __restrict__ p0re, const float* __restrict__ p0im,
    const float* __restrict__ p1re, const float* __restrict__ p1im,
    const float* __restrict__ p2re, const float* __restrict__ p2im,
    const float* __restrict__ p3re, const float* __restrict__ p3im,
    float* __restrict__ out, int npoints) {
  const int wave = (int)((blockIdx.x * blockDim.x + threadIdx.x) >> 5);  // wave-uniform
  const int lane = (int)(threadIdx.x & 31);
  if (wave >= npoints) return;  // uniform per wave: EXEC stays all-ones for live waves

  float* obase = out + (size_t)wave * OUT_STRIDE;

  // part0: complex64[2C] = (re0[0,c], 0), (im0[0,c], 0) interleaved -> float4 per channel
  if (lane < 16) {
    v4f v;
    v.x = p0re[(size_t)wave * NCH + lane];
    v.y = 0.0f;
    v.z = p0im[(size_t)wave * NCH + lane];
    v.w = 0.0f;
    __builtin_nontemporal_store(v, (v4f*)(obase + lane * 4));
  }

  // norms for l = 0..3 at complex offset 32 + l*256
  gram_l<1>(p0re + (size_t)wave * 1 * NCH, p0im + (size_t)wave * 1 * NCH,
            obase + (32 + 0 * 256) * 2, lane);
  gram_l<3>(p1re + (size_t)wave * 3 * NCH, p1im + (size_t)wave * 3 * NCH,
            obase + (32 + 1 * 256) * 2, lane);
  gram_l<5>(p2re + (size_t)wave * 5 * NCH, p2im + (size_t)wave * 5 * NCH,
            obase + (32 + 2 * 256) * 2, lane);
  gram_l<7>(p3re + (size_t)wave * 7 * NCH, p3im + (size_t)wave * 7 * NCH,
            obase + (32 + 3 * 256) * 2, lane);
}

extern "C" void kernel_launch(void* const* d_in, const int* in_sizes, int n_in,
                              void* d_out, int out_size, void* d_ws, size_t ws_size,
                              hipStream_t stream) {
  const float* p0re = (const float*)d_in[0];
  const float* p0im = (const float*)d_in[1];
  const float* p1re = (const float*)d_in[2];
  const float* p1im = (const float*)d_in[3];
  const float* p2re = (const float*)d_in[4];
  const float* p2im = (const float*)d_in[5];
  const float* p3re = (const float*)d_in[6];
  const float* p3im = (const float*)d_in[7];
  float* out = (float*)d_out;

  const int npoints = in_sizes[0] / NCH;  // B*N (p0_re is B*N*1*C)
  const int waves_per_block = 256 / 32;   // 8 points per block
  const int blocks = (npoints + waves_per_block - 1) / waves_per_block;

  so3_scalars_kernel<<<blocks, 256, 0, stream>>>(p0re, p0im, p1re, p1im, p2re, p2im,
                                                 p3re, p3im, out, npoints);
}